// VmapSphConvAttention_62053687492658
// MI455X (gfx1250) — compile-verified
//
#include <hip/hip_runtime.h>
#include <hip/hip_bf16.h>

// ---------------------------------------------------------------------------
// VmapSphConvAttention for MI455X (gfx1250, wave32)
//   K1: Q/K per-head GEMMs via V_WMMA_F32_16X16X4_F32 (fp32-exact)
//   K2: zero output
//   K3: per-edge wave32 pass: gather q/k (L2-resident), stream w_ij (NT),
//       3 head dots + wave reduction, 15 f32 atomic scatter-adds
// ---------------------------------------------------------------------------

#define N_NODES   10000
#define N_PAIRS   200000
#define NFEAT     384
#define NHEADS    3
#define F_HEAD    128
#define SPH_DIM   15
#define M_TILES   (N_NODES / 16)   // 625

typedef float v2f __attribute__((ext_vector_type(2)));
typedef float v4f __attribute__((ext_vector_type(4)));
typedef float v8f __attribute__((ext_vector_type(8)));

// ---------------------------------------------------------------------------
// Kernel 1: q/k = x_h @ W^T per head, 16x16 output tile per wave.
// grid = (625, 3, 2[q|k]); block = 256 (8 waves); wave w owns column tile w.
// A(16x4 f32, 2 VGPR): lane holds A[m=lane&15][kb], A[m][kb+1], kb=(lane>>4)*2
// B(4x16 f32, 2 VGPR): lane holds B[kb][n=lane&15], B[kb+1][n]
//   B[f][g] = W[g][f]  (nn.Linear stores (out,in), y = x @ W.T)
// D(16x16 f32, 8 VGPR): vgpr r, lane L -> row r + 8*(L>>4), col L&15
// ---------------------------------------------------------------------------
__global__ __launch_bounds__(256) void qk_gemm_kernel(
    const float* __restrict__ x,
    const float* __restrict__ Wq,
    const float* __restrict__ Wk,
    float* __restrict__ qout,
    float* __restrict__ kout)
{
    const int mt   = blockIdx.x;          // node tile: rows mt*16 .. mt*16+15
    const int h    = blockIdx.y;          // head
    const int qk   = blockIdx.z;          // 0 -> Q, 1 -> K
    const int wave = threadIdx.x >> 5;    // column tile 0..7
    const int lane = threadIdx.x & 31;

    const float* __restrict__ W   = (qk == 0 ? Wq : Wk) + (size_t)h * F_HEAD * F_HEAD;
    float*       __restrict__ out = (qk == 0 ? qout : kout);

    const int mrow  = mt * 16 + (lane & 15);
    const int ncol  = wave * 16 + (lane & 15);
    const int khalf = (lane >> 4) << 1;   // 0 or 2

    const float* aptr = x + (size_t)mrow * NFEAT + h * F_HEAD + khalf;
    const float* bptr = W + (size_t)ncol * F_HEAD + khalf;   // row g of W, stride 1 in f

    v8f acc = {};
#pragma unroll
    for (int kc = 0; kc < F_HEAD / 4; ++kc) {
        v2f a = *(const v2f*)(aptr + kc * 4);
        v2f b = *(const v2f*)(bptr + kc * 4);
        // (neg_a, A, neg_b, B, c_mod, C, reuse_a, reuse_b)
        acc = __builtin_amdgcn_wmma_f32_16x16x4_f32(
            false, a, false, b, (short)0, acc, false, false);
    }

    float* obase = out + (size_t)(mt * 16) * NFEAT + h * F_HEAD
                 + wave * 16 + (lane & 15);
    const int rowoff = (lane >> 4) * 8;
#pragma unroll
    for (int r = 0; r < 8; ++r)
        obase[(size_t)(r + rowoff) * NFEAT] = acc[r];
}

// ---------------------------------------------------------------------------
// Kernel 2: zero output accumulator
// ---------------------------------------------------------------------------
__global__ void zero_kernel(float* __restrict__ out, int n)
{
    int i = blockIdx.x * blockDim.x + threadIdx.x;
    if (i < n) out[i] = 0.0f;
}

// ---------------------------------------------------------------------------
// Kernel 3: one wave32 per edge.
//   alpha_h = dot(q[i,h,:], w[e,h,:] * k[j,h,:]) / sqrt(128) * (phi_r + phi_chi)
//   out[i, s] += alpha_{head(s)} * sph[e, s],   head map [3,5,7] over s=0..14
// Each lane covers 4 contiguous features per head (v4f = global_load_b128).
// w_ij is streamed with non-temporal loads so q/k stay hot in L2.
// ---------------------------------------------------------------------------
__global__ __launch_bounds__(256) void edge_kernel(
    const float* __restrict__ q,
    const float* __restrict__ k,
    const float* __restrict__ w_ij,
    const float* __restrict__ sph,
    const int*   __restrict__ idx_i,
    const int*   __restrict__ idx_j,
    const float* __restrict__ phi_r,
    const float* __restrict__ phi_chi,
    float* __restrict__ out)
{
    const int e    = (blockIdx.x * blockDim.x + threadIdx.x) >> 5;
    const int lane = threadIdx.x & 31;
    if (e >= N_PAIRS) return;

    const int i = idx_i[e];
    const int j = idx_j[e];
    const float scale = (phi_r[e] + phi_chi[e]) * 0.08838834764831845f; // 1/sqrt(128)

    const v4f* qe = (const v4f*)(q    + (size_t)i * NFEAT);
    const v4f* ke = (const v4f*)(k    + (size_t)j * NFEAT);
    const v4f* we = (const v4f*)(w_ij + (size_t)e * NFEAT);

    float alpha[NHEADS];
#pragma unroll
    for (int h = 0; h < NHEADS; ++h) {
        const int v = h * 32 + lane;           // v4f index within row
        v4f qa = qe[v];
        v4f ka = ke[v];
        v4f wa = __builtin_nontemporal_load(we + v);
        float p = qa.x * wa.x * ka.x
                + qa.y * wa.y * ka.y
                + qa.z * wa.z * ka.z
                + qa.w * wa.w * ka.w;
        // full wave32 xor-reduction: result broadcast to all lanes
#pragma unroll
        for (int off = 16; off > 0; off >>= 1)
            p += __shfl_xor(p, off, 32);
        alpha[h] = p * scale;
    }

    if (lane < SPH_DIM) {
        const int h = (lane < 3) ? 0 : (lane < 8 ? 1 : 2);
        float s = __builtin_nontemporal_load(sph + (size_t)e * SPH_DIM + lane);
        atomicAdd(&out[(size_t)i * SPH_DIM + lane], alpha[h] * s);
    }
}

// ---------------------------------------------------------------------------
// Host launcher
// setup_inputs order: chi(0), sph_ij(1), x(2), w_ij(3), idx_i(4),
//                     phi_r_cut(5), phi_chi_cut(6), idx_j(7), Wq(8), Wk(9)
// ---------------------------------------------------------------------------
extern "C" void kernel_launch(void* const* d_in, const int* in_sizes, int n_in,
                              void* d_out, int out_size, void* d_ws, size_t ws_size,
                              hipStream_t stream)
{
    const float* sph     = (const float*)d_in[1];
    const float* x       = (const float*)d_in[2];
    const float* w_ij    = (const float*)d_in[3];
    const int*   idx_i   = (const int*)  d_in[4];
    const float* phi_r   = (const float*)d_in[5];
    const float* phi_chi = (const float*)d_in[6];
    const int*   idx_j   = (const int*)  d_in[7];
    const float* Wq      = (const float*)d_in[8];
    const float* Wk      = (const float*)d_in[9];

    float* out = (float*)d_out;
    float* q   = (float*)d_ws;                      // 10000*384 f32
    float* kk  = q + (size_t)N_NODES * NFEAT;       // 10000*384 f32 (30.7 MB total)

    // K1: per-head Q/K GEMMs via fp32 WMMA
    dim3 g1(M_TILES, NHEADS, 2);
    qk_gemm_kernel<<<g1, 256, 0, stream>>>(x, Wq, Wk, q, kk);

    // K2: zero (10000 x 15) accumulator
    const int nOut = N_NODES * SPH_DIM;
    zero_kernel<<<(nOut + 255) / 256, 256, 0, stream>>>(out, nOut);

    // K3: edge pass, 8 edges (waves) per 256-thread block
    const int nWaveBlocks = (N_PAIRS + 7) / 8;
    edge_kernel<<<nWaveBlocks, 256, 0, stream>>>(q, kk, w_ij, sph,
                                                 idx_i, idx_j, phi_r, phi_chi, out);
}